// DeformConvBlock_56418690400601
// MI455X (gfx1250) — compile-verified
//
#include <hip/hip_runtime.h>
#include <hip/hip_bf16.h>
#include <stdint.h>

// ---------------------------------------------------------------------------
// Problem constants (from reference setup_inputs)
// ---------------------------------------------------------------------------
#define NB   8      // batch
#define CIN  128    // input channels
#define OUTC 128    // output channels
#define HH   80
#define WW   80
#define HW   (HH * WW)          // 6400
#define NPOS (NB * HW)          // 51200 spatial positions
#define NTAP 9                  // 3x3 taps
#define OFFC 18                 // offset channels

// GEMM tiling
#define CHUNK   16                  // channels per K chunk
#define NCHUNK  (CIN / CHUNK)       // 8
#define KCH     (CHUNK * NTAP)      // 144 K rows per chunk
#define PAIRS   (KCH / 2)           // 72 row-pairs per chunk
#define KSTEPS  (KCH / 4)           // 36 WMMA K-steps per chunk
#define WLDS_DW (PAIRS * OUTC * 2)  // 18432 dwords = 72KB weight stage
#define BLDS_DW (8 * PAIRS * 16 * 2)// 18432 dwords = 72KB B stage (8 waves)
#define CLDS_DW (8 * 144 * 8)       // 9216 dwords  = 36KB coords (8 waves)
#define GEMM_LDS_BYTES ((WLDS_DW + BLDS_DW + CLDS_DW) * 4)  // 184320

typedef __attribute__((ext_vector_type(2))) float v2f;
typedef __attribute__((ext_vector_type(4))) float v4f;
typedef __attribute__((ext_vector_type(8))) float v8f;
typedef __attribute__((ext_vector_type(4))) unsigned int u32x4;
typedef __attribute__((ext_vector_type(8))) int i32x8;
typedef __attribute__((ext_vector_type(4))) int i32x4;

// ---------------------------------------------------------------------------
// TDM: 1-D contiguous global -> LDS copy (nelem fp32 elements, nelem < 65536)
// D# layout per CDNA5 ISA (count=1, type=2,
// data_size=4B, tensor_dim0 = tile_dim0 = nelem, stride = nelem).
// amdgpu-toolchain (clang-23) 6-arg builtin form.
// ---------------------------------------------------------------------------
__device__ __forceinline__ void tdm_load_1d(const float* gsrc,
                                            unsigned lds_byte_off,
                                            unsigned nelem) {
  unsigned long long ga = (unsigned long long)(uintptr_t)gsrc;
  u32x4 g0;
  g0[0] = 1u;                                   // count = 1 (valid descriptor)
  g0[1] = lds_byte_off;                         // lds_addr
  g0[2] = (unsigned)(ga & 0xFFFFFFFFu);         // global_addr[31:0]
  g0[3] = (unsigned)((ga >> 32) & 0x01FFFFFFu)  // global_addr[56:32]
        | 0x80000000u;                          // type = 2 ("image")
  i32x8 g1;
  g1[0] = 0x00020000;                 // workgroup_mask=0, data_size=2 (4B)
  g1[1] = (int)(nelem << 16);         // tensor_dim0 low16  (bits 63:48)
  g1[2] = (int)(1u << 16);            // tensor_dim0 hi16=0, tensor_dim1=1
  g1[3] = (int)(nelem << 16);         // tensor_dim1 hi=0, tile_dim0 = nelem
  g1[4] = 0;                          // tile_dim1 = 0, tile_dim2 = 0
  g1[5] = (int)nelem;                 // tensor_dim0_stride low32
  g1[6] = 0;
  g1[7] = 0;
  i32x4 z4 = {0, 0, 0, 0};            // groups 2/3 unused (<=2-D tensor)
  i32x8 z8 = {0, 0, 0, 0, 0, 0, 0, 0};
  __builtin_amdgcn_tensor_load_to_lds(g0, g1, z4, z4, z8, 0);
}

// ---------------------------------------------------------------------------
// Kernel 0: x NCHW -> NHWC  (channels contiguous => coalesced gathers later)
// ---------------------------------------------------------------------------
__global__ __launch_bounds__(256) void k_transpose(const float* __restrict__ x,
                                                   float* __restrict__ xt) {
  int idx = blockIdx.x * 256 + threadIdx.x;      // coalesced read of NCHW
  int s = idx % HW;
  int t = idx / HW;
  int c = t % CIN;
  int n = t / CIN;
  xt[(n * HW + s) * CIN + c] = x[idx];
}

// ---------------------------------------------------------------------------
// Kernel 1: repack weights.
//  wp : main conv weights in WMMA-A pair-interleaved layout
//       wp[((cc*72 + q)*128 + oc)*2 + e]  where kk = cc*144 + 2q + e,
//       c = kk/9, tap = kk%9  (kk ordered c-major within chunk)
//  wop: offset-conv weights as [c*9+tap][18]
// ---------------------------------------------------------------------------
__global__ __launch_bounds__(256) void k_repack(const float* __restrict__ w,
                                                const float* __restrict__ w_off,
                                                float* __restrict__ wp,
                                                float* __restrict__ wop) {
  int i = blockIdx.x * 256 + threadIdx.x;
  if (i < OUTC * CIN * NTAP) {
    int e  = i & 1;
    int t  = i >> 1;
    int oc = t % OUTC;
    int qq = t / OUTC;            // global pair index 0..575
    int kk = 2 * qq + e;          // global K row = c*9 + tap
    int c  = kk / NTAP;
    int tap = kk % NTAP;
    wp[i] = w[(oc * CIN + c) * NTAP + tap];
  }
  if (i < CIN * NTAP * OFFC) {
    int o  = i % OFFC;
    int t2 = i / OFFC;
    int tap = t2 % NTAP;
    int c  = t2 / NTAP;
    wop[i] = w_off[(o * CIN + c) * NTAP + tap];
  }
}

// ---------------------------------------------------------------------------
// Kernel 2: offset conv (3x3, 18 out channels), zero padding.
// One thread per position, 18 accumulators; weights broadcast from LDS.
// Output layout [pos][18] (position-contiguous for the GEMM kernel).
// ---------------------------------------------------------------------------
__global__ __launch_bounds__(256) void k_offset_conv(const float* __restrict__ xt,
                                                     const float* __restrict__ wop,
                                                     const float* __restrict__ b_off,
                                                     float* __restrict__ offt) {
  extern __shared__ float sw[];                 // CIN*9*18 = 20736 dwords
  for (int i = threadIdx.x; i < CIN * NTAP * OFFC; i += 256) sw[i] = wop[i];
  __syncthreads();

  int p = blockIdx.x * 256 + threadIdx.x;       // 200 blocks * 256 = 51200
  int n = p / HW;
  int rem = p % HW;
  int y = rem / WW;
  int x = rem % WW;

  float acc[OFFC];
#pragma unroll
  for (int o = 0; o < OFFC; ++o) acc[o] = 0.0f;

  for (int tap = 0; tap < NTAP; ++tap) {
    int y2 = y + tap / 3 - 1;
    int x2 = x + tap % 3 - 1;
    if ((unsigned)y2 < (unsigned)HH && (unsigned)x2 < (unsigned)WW) {
      const float* xp = xt + ((n * HH + y2) * WW + x2) * CIN;
      for (int c = 0; c < CIN; ++c) {
        float v = xp[c];
        const float* wr = sw + (c * NTAP + tap) * OFFC;   // 72B-aligned row
#pragma unroll
        for (int o = 0; o < OFFC; o += 2) {
          v2f w2 = *(const v2f*)&wr[o];                    // ds_load_b64
          acc[o]     += v * w2.x;
          acc[o + 1] += v * w2.y;
        }
      }
    }
  }
  float* op = offt + p * OFFC;
#pragma unroll
  for (int o = 0; o < OFFC; ++o) op[o] = acc[o] + b_off[o];
}

// ---------------------------------------------------------------------------
// Kernel 3: fused bilinear sampling + implicit GEMM via V_WMMA_F32_16X16X4_F32.
// Wave tile: M=128 (8 x 16x16 f32 accumulators), N=16 positions, K chunked.
// Weights staged to LDS by the TDM; B built by the wave in WMMA layout.
// ---------------------------------------------------------------------------
__global__ __launch_bounds__(256) void k_deform_gemm(const float* __restrict__ xt,
                                                     const float* __restrict__ offt,
                                                     const float* __restrict__ wp,
                                                     const float* __restrict__ bias,
                                                     float* __restrict__ out) {
  extern __shared__ float smem[];
  float* wlds = smem;                       // [72 pairs][128 oc][2]  (72KB)
  float* blds = smem + WLDS_DW;             // per wave: [72][16][2]  (72KB)
  float* clds = smem + WLDS_DW + BLDS_DW;   // per wave: [144][8]     (36KB)

  const int tid  = threadIdx.x;
  const int lane = tid & 31;
  const int wave = tid >> 5;
  const int hi   = lane >> 4;               // half-wave select
  const int col  = lane & 15;

  const int pbase = blockIdx.x * 128 + wave * 16;   // 16 consecutive positions
  const int n   = pbase / HW;
  const int rem = pbase % HW;
  const int y   = rem / WW;
  const int x0  = rem % WW;                 // tile stays inside one row

  float* myb = blds + wave * (PAIRS * 16 * 2);
  float* myc = clds + wave * (144 * 8);

  // --- precompute bilinear coords: 144 (tap,pos) sets, masks folded in ---
  for (int i = lane; i < 144; i += 32) {
    int tap = i >> 4, pos = i & 15;
    const float* op = offt + (size_t)(pbase + pos) * OFFC;
    float dy = op[2 * tap];
    float dx = op[2 * tap + 1];
    float py = (float)(y + tap / 3 - 1) + dy;
    float px = (float)(x0 + pos + tap % 3 - 1) + dx;
    float fy = floorf(py), fx = floorf(px);
    float ay = py - fy, ax = px - fx;
    int y0 = (int)fy, xq = (int)fx;
    int idx[4];
    float wt[4];
#pragma unroll
    for (int cr = 0; cr < 4; ++cr) {
      int cy = y0 + (cr >> 1);
      int cx = xq + (cr & 1);
      bool ok = (unsigned)cy < (unsigned)HH && (unsigned)cx < (unsigned)WW;
      float wy = (cr >> 1) ? ay : (1.0f - ay);
      float wx = (cr & 1) ? ax : (1.0f - ax);
      idx[cr] = ok ? ((n * HH + cy) * WW + cx) * CIN : 0;
      wt[cr]  = ok ? wy * wx : 0.0f;
    }
    v4f iv = { __int_as_float(idx[0]), __int_as_float(idx[1]),
               __int_as_float(idx[2]), __int_as_float(idx[3]) };
    v4f wv = { wt[0], wt[1], wt[2], wt[3] };
    ((v4f*)myc)[i * 2]     = iv;
    ((v4f*)myc)[i * 2 + 1] = wv;
  }
  // coords are produced & consumed by the same wave: LDS ops are in-order.

  v8f acc[8];
#pragma unroll
  for (int m = 0; m < 8; ++m) acc[m] = (v8f){0, 0, 0, 0, 0, 0, 0, 0};

  const unsigned wlds_off = (unsigned)(uintptr_t)(void*)wlds;

  for (int cc = 0; cc < NCHUNK; ++cc) {
    __syncthreads();                               // wlds safe to overwrite
    if (tid == 0)                                  // one wave drives the TDM
      tdm_load_1d(wp + (size_t)cc * WLDS_DW, wlds_off, WLDS_DW);

    // --- bilinear sampling for this channel chunk (overlaps TDM) ---
    // lanes 0-15: channels cc*16+0..15 of even sets; 16-31: odd sets.
    const float* xc = xt + cc * CHUNK + col;
    for (int t = 0; t < 72; ++t) {
      int set = 2 * t + hi;
      v4f iv = ((const v4f*)myc)[set * 2];
      v4f wv = ((const v4f*)myc)[set * 2 + 1];
      float v = wv.x * xc[__float_as_int(iv.x)]
              + wv.y * xc[__float_as_int(iv.y)]
              + wv.z * xc[__float_as_int(iv.z)]
              + wv.w * xc[__float_as_int(iv.w)];
      int tap = set >> 4, pos = set & 15;
      int kk = col * NTAP + tap;                   // K row (matches wp order)
      myb[((kk >> 1) * 16 + pos) * 2 + (kk & 1)] = v;
    }

    if (tid == 0) __builtin_amdgcn_s_wait_tensorcnt(0);
    __syncthreads();                               // weights visible to all

    // --- WMMA: 36 K-steps x 8 M-tiles, fp32 matrix pipe ---
    for (int s = 0; s < KSTEPS; ++s) {
      v2f bfrag = *(const v2f*)&myb[((2 * s + hi) * 16 + col) * 2];
#pragma unroll
      for (int m = 0; m < 8; ++m) {
        v2f afrag = *(const v2f*)&wlds[((2 * s + hi) * OUTC + m * 16 + col) * 2];
        acc[m] = __builtin_amdgcn_wmma_f32_16x16x4_f32(
            false, afrag, false, bfrag, (short)0, acc[m], false, false);
      }
    }
  }

  // --- epilogue: add bias, store NCHW ---
  const size_t obase = (size_t)n * OUTC * HW + (size_t)y * WW + x0 + col;
#pragma unroll
  for (int m = 0; m < 8; ++m) {
#pragma unroll
    for (int r = 0; r < 8; ++r) {
      int oc = m * 16 + r + hi * 8;                // C/D VGPR layout
      out[obase + (size_t)oc * HW] = acc[m][r] + bias[oc];
    }
  }
}

// ---------------------------------------------------------------------------
// Host launcher
// ---------------------------------------------------------------------------
extern "C" void kernel_launch(void* const* d_in, const int* in_sizes, int n_in,
                              void* d_out, int out_size, void* d_ws, size_t ws_size,
                              hipStream_t stream) {
  const float* x     = (const float*)d_in[0];   // [8,128,80,80]
  const float* w_off = (const float*)d_in[1];   // [18,128,3,3]
  const float* b_off = (const float*)d_in[2];   // [18]
  const float* w     = (const float*)d_in[3];   // [128,128,3,3]
  const float* b     = (const float*)d_in[4];   // [128]
  float* out = (float*)d_out;                   // [8,128,80,80]

  float* ws   = (float*)d_ws;
  float* xt   = ws;                              // 6,553,600 f
  float* offt = xt + (size_t)NB * CIN * HW;      //   921,600 f
  float* wp   = offt + (size_t)NPOS * OFFC;      //   147,456 f
  float* wop  = wp + (size_t)OUTC * CIN * NTAP;  //    20,736 f

  k_transpose<<<(NB * CIN * HW) / 256, 256, 0, stream>>>(x, xt);
  k_repack<<<(OUTC * CIN * NTAP + 255) / 256, 256, 0, stream>>>(w, w_off, wp, wop);
  k_offset_conv<<<NPOS / 256, 256, CIN * NTAP * OFFC * 4, stream>>>(xt, wop, b_off, offt);
  k_deform_gemm<<<NPOS / 128, 256, GEMM_LDS_BYTES, stream>>>(xt, offt, wp, b, out);
}